// Routing_2396591751876
// MI455X (gfx1250) — compile-verified
//
#include <hip/hip_runtime.h>
#include <hip/hip_bf16.h>

typedef float v2f __attribute__((ext_vector_type(2)));
typedef float v8f __attribute__((ext_vector_type(8)));

#define MAX_LEN 200
#define MAX_K   3
#define D_IN    128
#define D_OUT   128
#define NBATCH  1024
#define NEG_PAD (-65535.0f)

// ---------------------------------------------------------------------------
// Kernel 1: low_new[b,l,e] = sum_d low[b,l,d] * S[d,e]
// Flat M = B*L = 204800 rows (12800 M-tiles of 16), N = 128 (8 N-tiles), K=128.
// One wave per 16x16 output tile, 32 WMMA steps of K=4 (f32 WMMA).
// A 16x4 f32 frag: lanes 0-15 -> M=lane, VGPR0/1 = K=0,1 ; lanes 16-31 -> K=2,3.
// ---------------------------------------------------------------------------
__global__ void __launch_bounds__(128)
gemm_lowS_kernel(const float* __restrict__ low, const float* __restrict__ S,
                 float* __restrict__ lown) {
  const int wave = threadIdx.x >> 5;
  const int lane = threadIdx.x & 31;
  const int tile = blockIdx.x * 4 + wave;          // 102400 tiles total
  const int tm   = tile >> 3;                      // 0..12799
  const int tn   = tile & 7;                       // 0..7

  const int m     = lane & 15;
  const int khalf = (lane >> 4) * 2;               // 0 or 2
  const float* __restrict__ Arow = low + (size_t)(tm * 16 + m) * D_IN;
  const int n = tn * 16 + (lane & 15);

  v8f c = {};
#pragma unroll
  for (int kk = 0; kk < 32; ++kk) {
    const int kb = kk * 4 + khalf;
    v2f a;  a.x = Arow[kb];          a.y = Arow[kb + 1];
    v2f bb; bb.x = S[kb * D_OUT + n]; bb.y = S[(kb + 1) * D_OUT + n];
    c = __builtin_amdgcn_wmma_f32_16x16x4_f32(false, a, false, bb,
                                              (short)0, c, false, false);
  }

  // C/D layout: lanes 0-15 -> N=lane, M=r ; lanes 16-31 -> N=lane-16, M=r+8
  const int cn    = tn * 16 + (lane & 15);
  const int rbase = tm * 16 + ((lane >> 4) << 3);
#pragma unroll
  for (int r = 0; r < 8; ++r)
    lown[(size_t)(rbase + r) * D_OUT + cn] = c[r];
}

// ---------------------------------------------------------------------------
// Kernel 2: init working B and zero delta accumulator
// ---------------------------------------------------------------------------
__global__ void init_b_kernel(const float* __restrict__ Bin,
                              float* __restrict__ Bwork,
                              float* __restrict__ Bdelta) {
  const int t = blockIdx.x * blockDim.x + threadIdx.x;
  if (t < MAX_K * MAX_LEN) { Bwork[t] = Bin[t]; Bdelta[t] = 0.0f; }
}

// ---------------------------------------------------------------------------
// Kernel 3: one routing iteration. One block (128 threads = 4 waves) per batch.
//  - masked softmax of B_work rows (per-batch mask from seq_len)
//  - high_raw = W[3x200] @ low_new[b][200x128] via f32 WMMA (M padded to 16)
//  - squash across K=3
//  - if !last: atomicAdd partial B_delta ; if last: write output
// ---------------------------------------------------------------------------
__global__ void __launch_bounds__(128)
routing_iter_kernel(const float* __restrict__ lown, const float* __restrict__ Bwork,
                    const int* __restrict__ seq_len, float* __restrict__ Bdelta,
                    float* __restrict__ out, int last) {
  __shared__ float sW[MAX_K][MAX_LEN];   // logits -> softmax weights
  __shared__ float sH[MAX_K][D_OUT];     // high (raw, then squashed)
  __shared__ float sred[128];

  const int b    = blockIdx.x;
  const int t    = threadIdx.x;
  const int wave = t >> 5;
  const int lane = t & 31;
  const int len  = seq_len[b];
  const float* __restrict__ lowb = lown + (size_t)b * MAX_LEN * D_OUT;

  // masked logits into LDS
  for (int i = t; i < MAX_K * MAX_LEN; i += 128) {
    const int l = i % MAX_LEN;
    sW[i / MAX_LEN][l] = (l < len) ? Bwork[i] : NEG_PAD;
  }
  __syncthreads();

  // softmax along L for each k (stable: subtract max)
  for (int k = 0; k < MAX_K; ++k) {
    float mx = -3.0e38f;
    for (int l = t; l < MAX_LEN; l += 128) mx = fmaxf(mx, sW[k][l]);
    sred[t] = mx; __syncthreads();
    for (int s = 64; s > 0; s >>= 1) {
      if (t < s) sred[t] = fmaxf(sred[t], sred[t + s]);
      __syncthreads();
    }
    const float mk = sred[0]; __syncthreads();
    float sum = 0.0f;
    for (int l = t; l < MAX_LEN; l += 128) {
      const float e = __expf(sW[k][l] - mk);
      sW[k][l] = e; sum += e;
    }
    sred[t] = sum; __syncthreads();
    for (int s = 64; s > 0; s >>= 1) {
      if (t < s) sred[t] += sred[t + s];
      __syncthreads();
    }
    const float inv = 1.0f / sred[0]; __syncthreads();
    for (int l = t; l < MAX_LEN; l += 128) sW[k][l] *= inv;
    __syncthreads();
  }

  // high_raw = W @ low_new[b] : M=16 (rows 3..15 zero), K=200 (50 x K4), N=128
  // each wave covers N-tiles {2w, 2w+1}
  const int m     = lane & 15;
  const int khalf = (lane >> 4) * 2;
#pragma unroll
  for (int j = 0; j < 2; ++j) {
    const int nt = wave * 2 + j;
    const int n  = nt * 16 + (lane & 15);
    v8f c = {};
    for (int kk = 0; kk < 50; ++kk) {
      const int kb = kk * 4 + khalf;
      v2f a;
      a.x = (m < MAX_K) ? sW[m][kb]     : 0.0f;
      a.y = (m < MAX_K) ? sW[m][kb + 1] : 0.0f;
      v2f bb;
      bb.x = lowb[kb * D_OUT + n];
      bb.y = lowb[(kb + 1) * D_OUT + n];
      c = __builtin_amdgcn_wmma_f32_16x16x4_f32(false, a, false, bb,
                                                (short)0, c, false, false);
    }
    if (lane < 16) {          // rows M=0..7 live in lanes 0-15; keep M=0..2
#pragma unroll
      for (int r = 0; r < MAX_K; ++r) sH[r][nt * 16 + lane] = c[r];
    }
  }
  __syncthreads();

  // squash across k (axis=1 of [B,K,D]) per (b,e)
  {
    const float x0 = sH[0][t], x1 = sH[1][t], x2 = sH[2][t];
    const float sq = x0 * x0 + x1 * x1 + x2 * x2;
    const float scale = sq / (1.0f + sq) * rsqrtf(sq + 1e-9f);
    sH[0][t] = scale * x0; sH[1][t] = scale * x1; sH[2][t] = scale * x2;
    if (last) {
      float* __restrict__ ob = out + (size_t)b * MAX_K * D_OUT;
      ob[0 * D_OUT + t] = sH[0][t];
      ob[1 * D_OUT + t] = sH[1][t];
      ob[2 * D_OUT + t] = sH[2][t];
    }
  }
  __syncthreads();

  // partial B_delta[k,l] = sum_e high[k,e] * low_new[b,l,e]  (L2-resident reads)
  if (!last) {
    for (int p = t; p < MAX_K * MAX_LEN; p += 128) {
      const int k = p / MAX_LEN;
      const int l = p % MAX_LEN;
      const float4* __restrict__ lp =
          reinterpret_cast<const float4*>(lowb + (size_t)l * D_OUT);
      float acc = 0.0f;
#pragma unroll
      for (int e4 = 0; e4 < D_OUT / 4; ++e4) {
        const float4 v = lp[e4];
        acc += v.x * sH[k][4 * e4 + 0];
        acc += v.y * sH[k][4 * e4 + 1];
        acc += v.z * sH[k][4 * e4 + 2];
        acc += v.w * sH[k][4 * e4 + 3];
      }
      atomicAdd(&Bdelta[p], acc);
    }
  }
}

// ---------------------------------------------------------------------------
// Kernel 4: B_work += B_delta ; B_delta = 0
// ---------------------------------------------------------------------------
__global__ void update_b_kernel(float* __restrict__ Bwork,
                                float* __restrict__ Bdelta) {
  const int t = blockIdx.x * blockDim.x + threadIdx.x;
  if (t < MAX_K * MAX_LEN) { Bwork[t] += Bdelta[t]; Bdelta[t] = 0.0f; }
}

extern "C" void kernel_launch(void* const* d_in, const int* in_sizes, int n_in,
                              void* d_out, int out_size, void* d_ws, size_t ws_size,
                              hipStream_t stream) {
  const float* low  = (const float*)d_in[0];   // [1024,200,128]
  const float* Bin  = (const float*)d_in[1];   // [1,3,200]
  const float* S    = (const float*)d_in[2];   // [128,128]
  const int*   slen = (const int*)d_in[3];     // [1024,1]
  float* out = (float*)d_out;                  // [1024,3,128]

  // workspace layout
  char* ws = (char*)d_ws;
  float* lown   = (float*)ws;                                   // 104,857,600 B
  float* Bwork  = (float*)(ws + (size_t)NBATCH * MAX_LEN * D_OUT * sizeof(float));
  float* Bdelta = Bwork + MAX_K * MAX_LEN;

  // K1: low_new = low @ S  (12800 M-tiles * 8 N-tiles, 4 waves/block)
  gemm_lowS_kernel<<<(12800 * 8) / 4, 128, 0, stream>>>(low, S, lown);

  // K2: init working B / delta accumulator
  init_b_kernel<<<5, 128, 0, stream>>>(Bin, Bwork, Bdelta);

  // 3 routing iterations
  for (int it = 0; it < 3; ++it) {
    const int last = (it == 2);
    routing_iter_kernel<<<NBATCH, 128, 0, stream>>>(lown, Bwork, slen, Bdelta,
                                                    out, last);
    if (!last) update_b_kernel<<<5, 128, 0, stream>>>(Bwork, Bdelta);
  }
}